// SpatialBlock_21251498181392
// MI455X (gfx1250) — compile-verified
//
#include <hip/hip_runtime.h>
#include <hip/hip_bf16.h>

// ---------------------------------------------------------------------------
// GATConv(+edge attr) + segment softmax + mean-head + LayerNorm + ReLU
// N=100000 nodes, E=1.6M edges, IN_C=128, OUT_C=64, H=2.
//
//  K0 zero     : accum / a_src / a_dst / amax-keys / denom
//  K1 gemm     : h = x @ W   (v_wmma_f32_16x16x4_f32; A tile staged into LDS
//                via GLOBAL_LOAD_ASYNC_TO_LDS_B128 + s_wait_asynccnt, read
//                back with ds_load_b64; bank-conflict-free padded stride)
//  K2 nodeatt  : a_src[n,h], a_dst[n,h]  (wave32 shfl reductions)
//  K3 edgeM    : M[4][2] = reduce(W_edge, att_edge)  (a_edge = edge_attr @ M)
//  K4 alpha    : alpha = lrelu(...); atomicMax monotonic-key (segment max)
//  K5 ex       : ex = exp(alpha - amax[dst]); atomicAdd denom[dst]
//  K6 scatter  : accum[dst] += h[src] * ex/denom  (1 wave/edge; accum+h fit
//                in 192MB L2 so f32 atomics run at L2 rate)
//  K7 finalize : head-mean + bias + LayerNorm + ReLU
//
// Workspace (~118.4 MB): h[N*128] | accum[N*128] | a_src[N*2] | a_dst[N*2]
//                        | amax(u32)[N*2] | denom[N*2] | exw[2*E] | M[8]
// ---------------------------------------------------------------------------

#define IN_C   128
#define OUT_C  64
#define NHEAD  2
#define NEG_SLOPE 0.2f
#define LN_EPS 1e-5f

typedef float v2f __attribute__((ext_vector_type(2)));
typedef float v8f __attribute__((ext_vector_type(8)));
typedef int   v4i __attribute__((ext_vector_type(4)));

#define GLOBAL_AS __attribute__((address_space(1)))
#define LDS_AS    __attribute__((address_space(3)))

#if __has_builtin(__builtin_amdgcn_global_load_async_to_lds_b128)
#define HAVE_ASYNC_LDS 1
#else
#define HAVE_ASYNC_LDS 0
#endif

// monotonic float <-> uint key (total order matches float order on finite vals)
__device__ __forceinline__ unsigned f2key(float f) {
  unsigned u = __float_as_uint(f);
  return (u & 0x80000000u) ? ~u : (u | 0x80000000u);
}
__device__ __forceinline__ float key2f(unsigned k) {
  unsigned u = (k & 0x80000000u) ? (k & 0x7FFFFFFFu) : ~k;
  return __uint_as_float(u);
}

// ---- K0: zero a contiguous word region ------------------------------------
__global__ void k_zero(unsigned* __restrict__ p, size_t n) {
  size_t i = (size_t)blockIdx.x * blockDim.x + threadIdx.x;
  if (i < n) p[i] = 0u;
}

// ---- K1: h = x @ W via V_WMMA_F32_16X16X4_F32 ------------------------------
// Block = 256 threads (8 waves) computes a 16-row x 128-col slab; wave w owns
// column tile n0 = 16*w.  A fragments are identical across the 8 waves, so the
// 16x128 A tile is staged once into LDS (async DMA when available), padded to
// stride 132 floats: row bank offset advances by 4 -> 16 distinct banks,
// 528B row stride keeps b128 stores 16B-aligned and b64 reads 8B-aligned.
// Requires gridDim.x == N/16 with N % 16 == 0 (holds: 100000 = 6250*16).
//
// 32-bit A (16x4): lane l(0..15): v0=A[l][k+0], v1=A[l][k+1]; lanes 16..31: K+2,K+3
// 32-bit B (4x16): lane l(0..15): v0=B[k+0][l], v1=B[k+1][l]; lanes 16..31: K+2,K+3
// C/D (16x16 f32): VGPR r, lane l: row = r + 8*(l>=16), col = l&15
#define A_STRIDE 132
__global__ void k_gemm_wmma(const float* __restrict__ x, const float* __restrict__ W,
                            float* __restrict__ h, int nNodes) {
  __shared__ float Al[16 * A_STRIDE];            // 8448 B

  const int wave = threadIdx.x >> 5;             // 0..7 => column tile
  const int lane = threadIdx.x & 31;
  const int m0 = blockIdx.x * 16;
  const int n0 = wave * 16;

  // ---- stage A tile (16 rows x 128 floats = 512 x 16B chunks) into LDS ----
  {
    const int t = threadIdx.x;
#pragma unroll
    for (int i = 0; i < 2; ++i) {
      const int c = t + i * 256;                 // chunk 0..511
      const int r = c >> 5;                      // row 0..15
      const int q = c & 31;                      // 16B chunk within row
      const float* gsrc = x + (size_t)(m0 + r) * IN_C + q * 4;
      float* ldst = &Al[r * A_STRIDE + q * 4];
#if HAVE_ASYNC_LDS
      __builtin_amdgcn_global_load_async_to_lds_b128(
          (GLOBAL_AS v4i*)gsrc, (LDS_AS v4i*)ldst, 0, 0);
#else
      *reinterpret_cast<float4*>(ldst) = *reinterpret_cast<const float4*>(gsrc);
#endif
    }
  }
#if HAVE_ASYNC_LDS
#if __has_builtin(__builtin_amdgcn_s_wait_asynccnt)
  __builtin_amdgcn_s_wait_asynccnt(0);
#else
  asm volatile("s_wait_asynccnt 0" ::: "memory");
#endif
#endif
  __syncthreads();

  // ---- WMMA main loop -----------------------------------------------------
  const int row   = lane & 15;
  const int khalf = (lane >> 4) << 1;            // 0 or 2
  const float* arow = &Al[row * A_STRIDE + khalf];

  v8f c = {};
#pragma unroll 4
  for (int k = 0; k < IN_C; k += 4) {
    const float2 av = *reinterpret_cast<const float2*>(arow + k);  // ds_load_b64
    v2f a; a.x = av.x; a.y = av.y;
    v2f b;
    b.x = W[(size_t)(k + khalf)     * IN_C + n0 + row];
    b.y = W[(size_t)(k + khalf + 1) * IN_C + n0 + row];
    c = __builtin_amdgcn_wmma_f32_16x16x4_f32(false, a, false, b,
                                              (short)0, c, false, false);
  }

  const int col   = n0 + (lane & 15);
  const int mbase = m0 + ((lane >> 4) << 3);
#pragma unroll
  for (int r = 0; r < 8; ++r)
    h[(size_t)(mbase + r) * IN_C + col] = c[r];
}

// ---- K2: per-node attention coefficients (1 wave / node) -------------------
__global__ void k_nodeatt(const float* __restrict__ h,
                          const float* __restrict__ att_src,
                          const float* __restrict__ att_dst,
                          float* __restrict__ asrc, float* __restrict__ adst,
                          int nNodes) {
  const int lane = threadIdx.x & 31;
  const int n = blockIdx.x * (blockDim.x >> 5) + (threadIdx.x >> 5);
  if (n >= nNodes) return;
  const float4 hv = reinterpret_cast<const float4*>(h + (size_t)n * IN_C)[lane];
  const float4 as = reinterpret_cast<const float4*>(att_src)[lane];  // flat [H*C]
  const float4 ad = reinterpret_cast<const float4*>(att_dst)[lane];
  float ss = hv.x*as.x + hv.y*as.y + hv.z*as.z + hv.w*as.w;
  float sd = hv.x*ad.x + hv.y*ad.y + hv.z*ad.z + hv.w*ad.w;
  // reduce within each 16-lane half (lanes 0-15 = head0, 16-31 = head1)
#pragma unroll
  for (int o = 8; o >= 1; o >>= 1) {
    ss += __shfl_xor(ss, o, 32);
    sd += __shfl_xor(sd, o, 32);
  }
  if ((lane & 15) == 0) {
    const int head = lane >> 4;
    asrc[(size_t)n * NHEAD + head] = ss;
    adst[(size_t)n * NHEAD + head] = sd;
  }
}

// ---- K3: fold att_edge into W_edge: M[d][head] -----------------------------
__global__ void k_edgeM(const float* __restrict__ W_edge,
                        const float* __restrict__ att_edge,
                        float* __restrict__ M) {
  const int t = threadIdx.x;
  if (t < 8) {
    const int d = t >> 1, head = t & 1;
    float s = 0.f;
    for (int c = 0; c < OUT_C; ++c)
      s += W_edge[d * (NHEAD*OUT_C) + head * OUT_C + c] * att_edge[head * OUT_C + c];
    M[d * NHEAD + head] = s;
  }
}

// ---- K4: alpha + segment max (keys) ----------------------------------------
__global__ void k_alpha(const long long* __restrict__ ei,
                        const float* __restrict__ ea,
                        const float* __restrict__ asrc, const float* __restrict__ adst,
                        const float* __restrict__ M,
                        float* __restrict__ exw, unsigned* __restrict__ amax,
                        long long E) {
  const long long e = (long long)blockIdx.x * blockDim.x + threadIdx.x;
  if (e >= E) return;
  const long long s = ei[e], d = ei[E + e];
  const float4 a = reinterpret_cast<const float4*>(ea)[e];
#pragma unroll
  for (int hh = 0; hh < NHEAD; ++hh) {
    float ae = a.x*M[0*NHEAD+hh] + a.y*M[1*NHEAD+hh] + a.z*M[2*NHEAD+hh] + a.w*M[3*NHEAD+hh];
    float al = asrc[s*NHEAD+hh] + adst[d*NHEAD+hh] + ae;
    al = (al > 0.f) ? al : NEG_SLOPE * al;          // leaky relu
    exw[(size_t)hh * E + e] = al;
    atomicMax(&amax[d*NHEAD+hh], f2key(al));
  }
}

// ---- K5: exp + segment sum -------------------------------------------------
__global__ void k_ex(const long long* __restrict__ ei,
                     const unsigned* __restrict__ amax,
                     float* __restrict__ exw, float* __restrict__ denom,
                     long long E) {
  const long long e = (long long)blockIdx.x * blockDim.x + threadIdx.x;
  if (e >= E) return;
  const long long d = ei[E + e];
#pragma unroll
  for (int hh = 0; hh < NHEAD; ++hh) {
    const float al = exw[(size_t)hh * E + e];
    const float mx = key2f(amax[d*NHEAD+hh]);
    const float ex = __expf(al - mx);
    exw[(size_t)hh * E + e] = ex;
    atomicAdd(&denom[d*NHEAD+hh], ex);
  }
}

// ---- K6: weighted message scatter (1 wave / edge) --------------------------
__global__ void k_scatter(const long long* __restrict__ ei,
                          const float* __restrict__ h,
                          const float* __restrict__ exw,
                          const float* __restrict__ denom,
                          float* __restrict__ accum, long long E) {
  const long long gid = (long long)blockIdx.x * blockDim.x + threadIdx.x;
  const long long e = gid >> 5;
  const int lane = threadIdx.x & 31;
  if (e >= E) return;
  const long long s = ei[e], d = ei[E + e];
  const float w0 = exw[e]     / (denom[d*NHEAD+0] + 1e-16f);
  const float w1 = exw[E + e] / (denom[d*NHEAD+1] + 1e-16f);
  const float w = (lane >= 16) ? w1 : w0;           // lanes 0-15: head0 chans
  const float4 hv = reinterpret_cast<const float4*>(h + (size_t)s * IN_C)[lane];
  float* ap = accum + (size_t)d * IN_C + lane * 4;
  atomicAdd(ap + 0, hv.x * w);
  atomicAdd(ap + 1, hv.y * w);
  atomicAdd(ap + 2, hv.z * w);
  atomicAdd(ap + 3, hv.w * w);
}

// ---- K7: head mean + bias + LayerNorm + ReLU (1 wave / node) ---------------
__global__ void k_finalize(const float* __restrict__ accum,
                           const float* __restrict__ bias,
                           const float* __restrict__ gamma,
                           const float* __restrict__ beta,
                           float* __restrict__ out, int nNodes) {
  const int lane = threadIdx.x & 31;
  const int n = blockIdx.x * (blockDim.x >> 5) + (threadIdx.x >> 5);
  if (n >= nNodes) return;
  const float* ac = accum + (size_t)n * IN_C;
  const int c0 = lane, c1 = lane + 32;
  float v0 = 0.5f * (ac[c0] + ac[OUT_C + c0]) + bias[c0];
  float v1 = 0.5f * (ac[c1] + ac[OUT_C + c1]) + bias[c1];
  float s = v0 + v1;
#pragma unroll
  for (int o = 16; o >= 1; o >>= 1) s += __shfl_xor(s, o, 32);
  const float mu = s * (1.0f / OUT_C);
  const float d0 = v0 - mu, d1 = v1 - mu;
  float q = d0*d0 + d1*d1;
#pragma unroll
  for (int o = 16; o >= 1; o >>= 1) q += __shfl_xor(q, o, 32);
  const float inv = rsqrtf(q * (1.0f / OUT_C) + LN_EPS);
  out[(size_t)n * OUT_C + c0] = fmaxf(d0 * inv * gamma[c0] + beta[c0], 0.f);
  out[(size_t)n * OUT_C + c1] = fmaxf(d1 * inv * gamma[c1] + beta[c1], 0.f);
}

// ---------------------------------------------------------------------------
extern "C" void kernel_launch(void* const* d_in, const int* in_sizes, int n_in,
                              void* d_out, int out_size, void* d_ws, size_t ws_size,
                              hipStream_t stream) {
  const float*     x        = (const float*)d_in[0];
  const long long* ei       = (const long long*)d_in[1];   // int64 [2,E]
  const float*     ea       = (const float*)d_in[2];
  const float*     W        = (const float*)d_in[3];
  const float*     att_src  = (const float*)d_in[4];
  const float*     att_dst  = (const float*)d_in[5];
  const float*     W_edge   = (const float*)d_in[6];
  const float*     att_edge = (const float*)d_in[7];
  const float*     bias     = (const float*)d_in[8];
  const float*     gamma    = (const float*)d_in[9];
  const float*     beta     = (const float*)d_in[10];
  float*           out      = (float*)d_out;

  const size_t    N = (size_t)in_sizes[0] / IN_C;      // 100000 (mult of 16)
  const long long E = (long long)in_sizes[1] / 2;      // 1600000

  // workspace carve-up
  float*    h     = (float*)d_ws;
  float*    accum = h + N * IN_C;
  float*    asrc  = accum + N * IN_C;
  float*    adst  = asrc + N * NHEAD;
  unsigned* amax  = (unsigned*)(adst + N * NHEAD);
  float*    denom = (float*)(amax + N * NHEAD);
  float*    exw   = denom + N * NHEAD;                  // [2, E]
  float*    Mbuf  = exw + (size_t)E * NHEAD;            // [4, 2]

  // K0: zero accum..denom (contiguous N*136 words); h/exw/M fully overwritten
  {
    const size_t nz = N * (IN_C + 4 * NHEAD);
    k_zero<<<(unsigned)((nz + 255) / 256), 256, 0, stream>>>((unsigned*)accum, nz);
  }
  // K1: WMMA GEMM (N multiple of 16; 8 waves/block cover all 128 columns)
  k_gemm_wmma<<<(unsigned)(N / 16), 256, 0, stream>>>(x, W, h, (int)N);
  // K2: node attention coefficients (8 waves/block = 8 nodes/block)
  k_nodeatt<<<(unsigned)((N + 7) / 8), 256, 0, stream>>>(h, att_src, att_dst,
                                                         asrc, adst, (int)N);
  // K3: fold edge attention matrix
  k_edgeM<<<1, 32, 0, stream>>>(W_edge, att_edge, Mbuf);
  // K4: alpha + segment max
  k_alpha<<<(unsigned)((E + 255) / 256), 256, 0, stream>>>(ei, ea, asrc, adst,
                                                           Mbuf, exw, amax, E);
  // K5: exp + segment sum
  k_ex<<<(unsigned)((E + 255) / 256), 256, 0, stream>>>(ei, amax, exw, denom, E);
  // K6: scatter messages (1 wave / edge)
  k_scatter<<<(unsigned)((E * 32 + 255) / 256), 256, 0, stream>>>(ei, h, exw,
                                                                  denom, accum, E);
  // K7: head-mean + LayerNorm + ReLU
  k_finalize<<<(unsigned)((N + 7) / 8), 256, 0, stream>>>(accum, bias, gamma,
                                                          beta, out, (int)N);
}